// VoConstruction_78142634983431
// MI455X (gfx1250) — compile-verified
//
#include <hip/hip_runtime.h>
#include <stdint.h>

// ---------------------------------------------------------------------------
// VoConstruction, closed form (delta = U[:,2] cancels: C d C^H = d):
//   c,s = cos/sin(theta=U[:,1]);  d1 = 2*x1;  d2 = 2*x2*(1-|x1|)
//   H00 = d1*c^2 + d2*s^2,  H11 = d1*s^2 + d2*c^2   (real)
//   H01 = c*s*(d2-d1)*e^{+2i psi},  H10 = conj(H01),  psi = U[:,0]
//   out = O @ H (row swap): row0 = (H10, H11), row1 = (H00, H01)
//
// Roofline: 52 B per element, ~218 MB total -> ~9.4 us floor @ 23.3 TB/s.
// Native __sincosf (v_sin_f32/v_cos_f32, TRANS pipe) keeps arithmetic
// intensity ~0.6 lane-ops/byte, far under machine balance -> memory-bound.
// CDNA5 paths: per-wave async global->LDS staging (ASYNCcnt double buffering)
// of the stride-12B U rows as coalesced b128 transactions; non-temporal b128
// stores for the write-once 134 MB output stream (don't pollute 192 MB L2).
// ---------------------------------------------------------------------------

typedef float v4f __attribute__((ext_vector_type(4)));

#define WAVE_TILE   128                      // elements per wave per stage
#define U_BYTES     (WAVE_TILE * 3 * 4)      // 1536
#define X_BYTES     (WAVE_TILE * 4)          // 512
#define BUF_BYTES   (U_BYTES + 2 * X_BYTES)  // 2560
#define WAVES_PB    8                        // 256 threads / wave32
#define TILES_PW    2                        // double-buffered pipeline depth

__device__ __forceinline__ uint32_t lds_addr_u32(const void* p) {
  // Flat LDS pointers carry the workgroup-relative LDS offset in addr[31:0]
  // (shared aperture occupies addr[63:32] on gfx1250).
  return (uint32_t)(uintptr_t)p;
}

__device__ __forceinline__ void async_b128(uint32_t lds_byte_addr,
                                           const void* gaddr) {
  asm volatile("global_load_async_to_lds_b128 %0, %1, off"
               :
               : "v"(lds_byte_addr), "v"(gaddr)
               : "memory");
}

__device__ __forceinline__ void wait_async_le5() {
  asm volatile("s_wait_asynccnt 5" ::: "memory");
}
__device__ __forceinline__ void wait_async_0() {
  asm volatile("s_wait_asynccnt 0" ::: "memory");
}

// Issue the 5 async b128 loads staging one 128-element tile into this wave's
// private LDS region. Caller guarantees a wave-uniform condition (full EXEC).
__device__ __forceinline__ void stage_tile(const char* Ub, const char* x1b,
                                           const char* x2b, char* lbuf,
                                           long tb, int lane) {
  const uint32_t l = lds_addr_u32(lbuf);
  const char* gU = Ub  + tb * 12 + lane * 16;
  const char* g1 = x1b + tb * 4  + lane * 16;
  const char* g2 = x2b + tb * 4  + lane * 16;
  async_b128(l + lane * 16 +    0, gU);
  async_b128(l + lane * 16 +  512, gU + 512);
  async_b128(l + lane * 16 + 1024, gU + 1024);
  async_b128(l + U_BYTES           + lane * 16, g1);
  async_b128(l + U_BYTES + X_BYTES + lane * 16, g2);
}

__device__ __forceinline__ void emit_elem(float psi, float theta, float l1,
                                          float l2raw, float* op) {
  float s, c;
  __sincosf(theta, &s, &c);                 // TRANS-pipe v_sin/v_cos
  float s2, c2;
  __sincosf(2.0f * psi, &s2, &c2);
  const float d1 = 2.0f * l1;
  const float d2 = 2.0f * l2raw * (1.0f - fabsf(l1));
  const float cc = c * c, ss = s * s;
  const float H00 = d1 * cc + d2 * ss;
  const float H11 = d1 * ss + d2 * cc;
  const float w   = c * s * (d2 - d1);
  v4f r0 = {w * c2, -w * s2, H11, 0.0f};    // (H10, H11)
  v4f r1 = {H00, 0.0f, w * c2, w * s2};     // (H00, H01)
  __builtin_nontemporal_store(r0, (v4f*)op);
  __builtin_nontemporal_store(r1, (v4f*)(op + 4));
}

__device__ __forceinline__ void compute_tile_lds(const char* lbuf, float* out,
                                                 long tb, int lane) {
  const float* uf = (const float*)(lbuf + lane * 48);  // 12 floats / lane
  const v4f a4 = *(const v4f*)(lbuf + U_BYTES + lane * 16);
  const v4f b4 = *(const v4f*)(lbuf + U_BYTES + X_BYTES + lane * 16);
  float* op = out + (tb + (long)lane * 4) * 8;
#pragma unroll
  for (int j = 0; j < 4; ++j)
    emit_elem(uf[3 * j], uf[3 * j + 1], a4[j], b4[j], op + j * 8);
}

// Guarded scalar fallback for a partial tail tile (dead for B = 4M).
__device__ __forceinline__ void compute_tile_direct(const float* U,
                                                    const float* x1,
                                                    const float* x2, float* out,
                                                    long tb, int lane, long n) {
#pragma unroll
  for (int j = 0; j < 4; ++j) {
    long e = tb + (long)lane * 4 + j;
    if (e < n)
      emit_elem(U[e * 3], U[e * 3 + 1], x1[e], x2[e], out + e * 8);
  }
}

__global__ __launch_bounds__(256) void vo_construction_kernel(
    const float* __restrict__ U, const float* __restrict__ x1,
    const float* __restrict__ x2, float* __restrict__ out, long n) {
  __shared__ __align__(16) char smem[WAVES_PB * TILES_PW * BUF_BYTES];

  const int lane = threadIdx.x & 31;
  const int wave = threadIdx.x >> 5;
  const long wid = (long)blockIdx.x * WAVES_PB + wave;  // global wave index
  char* buf0 = smem + wave * (TILES_PW * BUF_BYTES);
  char* buf1 = buf0 + BUF_BYTES;

  const long tb0 = (wid * TILES_PW + 0) * WAVE_TILE;
  const long tb1 = (wid * TILES_PW + 1) * WAVE_TILE;
  const bool full0 = (tb0 + WAVE_TILE) <= n;  // wave-uniform
  const bool full1 = (tb1 + WAVE_TILE) <= n;  // wave-uniform

  // Double-buffered async pipeline: issue both tiles, overlap wait/compute.
  if (full0) stage_tile((const char*)U, (const char*)x1, (const char*)x2,
                        buf0, tb0, lane);
  if (full1) stage_tile((const char*)U, (const char*)x1, (const char*)x2,
                        buf1, tb1, lane);

  if (full0) {
    if (full1) wait_async_le5();  // tile0's 5 loads done (in-order issue),
    else       wait_async_0();    // tile1 still in flight
    compute_tile_lds(buf0, out, tb0, lane);
  } else if (tb0 < n) {
    compute_tile_direct(U, x1, x2, out, tb0, lane, n);
  }

  if (full1) {
    wait_async_0();
    compute_tile_lds(buf1, out, tb1, lane);
  } else if (tb1 < n) {
    compute_tile_direct(U, x1, x2, out, tb1, lane, n);
  }
}

extern "C" void kernel_launch(void* const* d_in, const int* in_sizes, int n_in,
                              void* d_out, int out_size, void* d_ws,
                              size_t ws_size, hipStream_t stream) {
  const float* U  = (const float*)d_in[0];  // [B,3]
  const float* x1 = (const float*)d_in[1];  // [B]
  const float* x2 = (const float*)d_in[2];  // [B]
  // d_in[3] is O = [[0,1],[1,0]]: the row swap is baked into the kernel.
  float* out = (float*)d_out;               // [B,2,2] complex64 -> B*8 floats

  const long n = in_sizes[1];
  const long tiles = (n + WAVE_TILE - 1) / WAVE_TILE;
  const long waves = (tiles + TILES_PW - 1) / TILES_PW;
  long blocks = (waves + WAVES_PB - 1) / WAVES_PB;
  if (blocks < 1) blocks = 1;

  vo_construction_kernel<<<dim3((unsigned)blocks), dim3(256), 0, stream>>>(
      U, x1, x2, out, n);
}